// GlobalGraphGAT_7885559955601
// MI455X (gfx1250) — compile-verified
//
#include <hip/hip_runtime.h>
#include <math.h>

typedef __attribute__((ext_vector_type(2))) float v2f;
typedef __attribute__((ext_vector_type(8))) float v8f;
typedef __attribute__((ext_vector_type(4))) int   gat_v4i;

#define GAT_NEG_SLOPE 0.2f

// ---- gfx1250 async global->LDS path (guarded; fallback = plain LDS copy) ----
#if __has_builtin(__builtin_amdgcn_global_load_async_to_lds_b128)
#define GAT_ASYNC_B128 1
typedef __attribute__((address_space(1))) gat_v4i* gat_gp;   // global src (AS1)
typedef __attribute__((address_space(3))) gat_v4i* gat_lp;   // LDS dst (AS3)
#endif

#if __has_builtin(__builtin_amdgcn_s_wait_asynccnt)
#define GAT_WAIT_ASYNC() __builtin_amdgcn_s_wait_asynccnt(0)
#else
#define GAT_WAIT_ASYNC() asm volatile("s_wait_asynccnt 0x0" ::: "memory")
#endif

// ---- order-preserving float<->uint key (for atomicMax softmax-max pass) ----
__device__ __forceinline__ unsigned gat_enc(float f) {
    unsigned u = __float_as_uint(f);
    return (u & 0x80000000u) ? ~u : (u | 0x80000000u);
}
__device__ __forceinline__ float gat_dec(unsigned k) {
    unsigned u = (k & 0x80000000u) ? (k & 0x7FFFFFFFu) : ~k;
    return __uint_as_float(u);
}
__device__ __forceinline__ float gat_leaky(float v) {
    return v > 0.0f ? v : GAT_NEG_SLOPE * v;
}

// =====================================================================
// f32 WMMA GEMM: C[M,Ncols] = A[M,K] x B[K,Ncols], row-major.
// Block = 128 threads (4 wave32). grid = (M/16, Ncols/64).
// Block stages the K x 64 weight slice into LDS via the async
// global->LDS pipe (ASYNCcnt-tracked) when available, then wave w
// computes the 16x16 tile at column colBase+16w with
// V_WMMA_F32_16X16X4_F32, B-fragments fed from ds_load.
// A-frag (ISA 7.12.2): lanes 0-15 hold K=k0,k0+1; lanes 16-31 K=k0+2,k0+3.
// C/D: vgpr r = rows r (lanes 0-15) / r+8 (lanes 16-31), col = lane%16.
// =====================================================================
__global__ void gat_wmma_gemm_f32(const float* __restrict__ A,
                                  const float* __restrict__ B,
                                  float* __restrict__ C,
                                  int K, int Ncols) {
    extern __shared__ float sB[];            // [K][64] column slice of B
    const int colBase = blockIdx.y * 64;
    const int chunks  = K * 16;              // float4 chunks in the slice

#if defined(GAT_ASYNC_B128)
    for (int c = threadIdx.x; c < chunks; c += blockDim.x) {
        const int k = c >> 4, j = (c & 15) << 2;
        __builtin_amdgcn_global_load_async_to_lds_b128(
            (gat_gp)(B + (size_t)k * Ncols + colBase + j),
            (gat_lp)(sB + k * 64 + j), 0, 0);
    }
    GAT_WAIT_ASYNC();
#else
    for (int c = threadIdx.x; c < chunks; c += blockDim.x) {
        const int k = c >> 4, j = (c & 15) << 2;
        *(float4*)(sB + k * 64 + j) =
            *(const float4*)(B + (size_t)k * Ncols + colBase + j);
    }
#endif
    __syncthreads();

    const int lane = threadIdx.x & 31;
    const int wave = threadIdx.x >> 5;
    const int lr   = lane & 15;
    const int half = lane >> 4;

    const int arow  = blockIdx.x * 16 + lr;     // A row for this lane
    const int bcolL = wave * 16 + lr;           // column within LDS slice

    v8f acc = {};
    for (int k0 = 0; k0 < K; k0 += 4) {
        const int ka = k0 + 2 * half;
        v2f a, b;
        a.x = A[(size_t)arow * K + ka];
        a.y = A[(size_t)arow * K + ka + 1];
        b.x = sB[ka * 64 + bcolL];
        b.y = sB[(ka + 1) * 64 + bcolL];
        acc = __builtin_amdgcn_wmma_f32_16x16x4_f32(
                  /*neg_a=*/false, a, /*neg_b=*/false, b,
                  /*c_mod=*/(short)0, acc, /*reuse_a=*/false, /*reuse_b=*/false);
    }

    const int ccol  = colBase + bcolL;
    const int rbase = blockIdx.x * 16 + half * 8;
#pragma unroll
    for (int r = 0; r < 8; ++r) {
        C[(size_t)(rbase + r) * Ncols + ccol] = acc[r];
    }
}

// ---------------------------------------------------------------------
// Per-node attention logits: al_src[n,h] = <h[n,h,:], a_src[h,:]>.
// ---------------------------------------------------------------------
__global__ void gat_attn_logits(const float* __restrict__ h,
                                const float* __restrict__ a_src,
                                const float* __restrict__ a_dst,
                                float* __restrict__ al_src,
                                float* __restrict__ al_dst,
                                int N, int H, int C) {
    int i = blockIdx.x * blockDim.x + threadIdx.x;
    if (i >= N * H) return;
    int n = i / H, hd = i % H;
    const float4* hp = (const float4*)(h + (size_t)n * H * C + (size_t)hd * C);
    const float4* as = (const float4*)(a_src + (size_t)hd * C);
    const float4* ad = (const float4*)(a_dst + (size_t)hd * C);
    float ss = 0.f, sd = 0.f;
    for (int c = 0; c < C / 4; ++c) {
        float4 v = hp[c], s4 = as[c], d4 = ad[c];
        ss += v.x * s4.x + v.y * s4.y + v.z * s4.z + v.w * s4.w;
        sd += v.x * d4.x + v.y * d4.y + v.z * d4.z + v.w * d4.w;
    }
    al_src[i] = ss;
    al_dst[i] = sd;
}

__device__ __forceinline__ void gat_edge_sd(const int* __restrict__ ei,
                                            int E, int e, int& s, int& d) {
    if (e < E) { s = ei[e]; d = ei[E + e]; }
    else       { s = e - E; d = e - E; }     // self loop
}

// Pass 1: per-destination running max (uint-key atomicMax; memset0 == -inf).
__global__ void gat_edge_max(const int* __restrict__ ei, int E, int Etot,
                             const float* __restrict__ al_src,
                             const float* __restrict__ al_dst,
                             unsigned* __restrict__ mkey, int H) {
    int e = blockIdx.x * blockDim.x + threadIdx.x;
    if (e >= Etot) return;
    int s, d; gat_edge_sd(ei, E, e, s, d);
    for (int hd = 0; hd < H; ++hd) {
        float v = gat_leaky(al_src[s * H + hd] + al_dst[d * H + hd]);
        atomicMax(&mkey[d * H + hd], gat_enc(v));
    }
}

// Pass 2: softmax denominator.
__global__ void gat_edge_denom(const int* __restrict__ ei, int E, int Etot,
                               const float* __restrict__ al_src,
                               const float* __restrict__ al_dst,
                               const unsigned* __restrict__ mkey,
                               float* __restrict__ denom, int H) {
    int e = blockIdx.x * blockDim.x + threadIdx.x;
    if (e >= Etot) return;
    int s, d; gat_edge_sd(ei, E, e, s, d);
    for (int hd = 0; hd < H; ++hd) {
        float v  = gat_leaky(al_src[s * H + hd] + al_dst[d * H + hd]);
        float ex = __expf(v - gat_dec(mkey[d * H + hd]));
        atomicAdd(&denom[d * H + hd], ex);
    }
}

// Pass 3: out[dst,h,:] += alpha * h[src,h,:]. One thread per (edge,head),
// float4 gathers of the source row (recomputes e; al/mkey/denom are tiny
// and L2-resident, cheaper than a per-edge ex buffer).
__global__ void gat_edge_scatter(const int* __restrict__ ei, int E, int Etot,
                                 const float* __restrict__ al_src,
                                 const float* __restrict__ al_dst,
                                 const unsigned* __restrict__ mkey,
                                 const float* __restrict__ denom,
                                 const float* __restrict__ h,
                                 float* __restrict__ out, int H, int C) {
    int i = blockIdx.x * blockDim.x + threadIdx.x;
    if (i >= Etot * H) return;
    int e = i / H, hd = i % H;
    int s, d; gat_edge_sd(ei, E, e, s, d);
    float v     = gat_leaky(al_src[s * H + hd] + al_dst[d * H + hd]);
    float alpha = __expf(v - gat_dec(mkey[d * H + hd])) / denom[d * H + hd];
    const float4* hp = (const float4*)(h + (size_t)s * H * C + (size_t)hd * C);
    float*        op = out + (size_t)d * H * C + (size_t)hd * C;
    __builtin_prefetch(hp, 0, 3);   // global_prefetch_b8 on the gathered row
    for (int c4 = 0; c4 < C / 4; ++c4) {
        float4 hv = hp[c4];
        atomicAdd(&op[4 * c4 + 0], alpha * hv.x);
        atomicAdd(&op[4 * c4 + 1], alpha * hv.y);
        atomicAdd(&op[4 * c4 + 2], alpha * hv.z);
        atomicAdd(&op[4 * c4 + 3], alpha * hv.w);
    }
}

__global__ void gat_bias_relu(float* __restrict__ x, const float* __restrict__ b,
                              int total, int F) {
    int i = blockIdx.x * blockDim.x + threadIdx.x;
    if (i >= total) return;
    float v = x[i] + b[i % F];
    x[i] = v > 0.f ? v : 0.f;
}

__global__ void gat_bias_add(float* __restrict__ x, const float* __restrict__ b,
                             int total, int F) {
    int i = blockIdx.x * blockDim.x + threadIdx.x;
    if (i >= total) return;
    x[i] += b[i % F];
}

extern "C" void kernel_launch(void* const* d_in, const int* in_sizes, int n_in,
                              void* d_out, int out_size, void* d_ws, size_t ws_size,
                              hipStream_t stream) {
    (void)n_in; (void)out_size; (void)ws_size;
    const float* x   = (const float*)d_in[0];
    const int*   ei  = (const int*)d_in[1];   // [2,E] int32 (jax x64 disabled)
    const float* W1  = (const float*)d_in[2];
    const float* a1s = (const float*)d_in[3];
    const float* a1d = (const float*)d_in[4];
    const float* b1  = (const float*)d_in[5];
    const float* W2  = (const float*)d_in[6];
    const float* a2s = (const float*)d_in[7];
    const float* a2d = (const float*)d_in[8];
    const float* b2  = (const float*)d_in[9];

    const int IN = 128, F1 = 128, H1 = 4, C1 = 32, F2 = 64;
    const int N    = in_sizes[0] / IN;   // 50000 (multiple of 16)
    const int E    = in_sizes[1] / 2;    // 800000
    const int Etot = E + N;

    // ---- workspace layout (floats) ----
    float* ws = (float*)d_ws;
    size_t off = 0;
    float*    h1   = ws + off; off += (size_t)N * F1;
    float*    out1 = ws + off; off += (size_t)N * F1;   // becomes relu(conv1+b1)
    float*    al1s = ws + off; off += (size_t)N * H1;
    float*    al1d = ws + off; off += (size_t)N * H1;
    unsigned* mk1  = (unsigned*)(ws + off); off += (size_t)N * H1;
    float*    dn1  = ws + off; off += (size_t)N * H1;
    float*    h2   = ws + off; off += (size_t)N * F2;
    float*    al2s = ws + off; off += (size_t)N;
    float*    al2d = ws + off; off += (size_t)N;
    unsigned* mk2  = (unsigned*)(ws + off); off += (size_t)N;
    float*    dn2  = ws + off; off += (size_t)N;

    // zero accumulators (graph-capturable memset nodes)
    (void)hipMemsetAsync(out1, 0, (size_t)N * F1 * sizeof(float), stream);
    (void)hipMemsetAsync(mk1,  0, (size_t)N * H1 * sizeof(unsigned), stream);
    (void)hipMemsetAsync(dn1,  0, (size_t)N * H1 * sizeof(float), stream);
    (void)hipMemsetAsync(d_out,0, (size_t)N * F2 * sizeof(float), stream);
    (void)hipMemsetAsync(mk2,  0, (size_t)N * sizeof(unsigned), stream);
    (void)hipMemsetAsync(dn2,  0, (size_t)N * sizeof(float), stream);

    const int TB = 256;
    const size_t lds1 = (size_t)IN * 64 * sizeof(float);   // 32 KB
    const size_t lds2 = (size_t)F1 * 64 * sizeof(float);   // 32 KB

    // ============ Layer 1: GATConv(128 -> 4x32, concat) + ReLU ============
    gat_wmma_gemm_f32<<<dim3(N / 16, F1 / 64), 128, lds1, stream>>>(x, W1, h1, IN, F1);
    gat_attn_logits<<<(N * H1 + TB - 1) / TB, TB, 0, stream>>>(h1, a1s, a1d,
                                                               al1s, al1d, N, H1, C1);
    gat_edge_max    <<<(Etot + TB - 1) / TB, TB, 0, stream>>>(ei, E, Etot,
                                                              al1s, al1d, mk1, H1);
    gat_edge_denom  <<<(Etot + TB - 1) / TB, TB, 0, stream>>>(ei, E, Etot,
                                                              al1s, al1d, mk1, dn1, H1);
    gat_edge_scatter<<<(Etot * H1 + TB - 1) / TB, TB, 0, stream>>>(ei, E, Etot,
                                                                   al1s, al1d, mk1, dn1,
                                                                   h1, out1, H1, C1);
    gat_bias_relu<<<((size_t)N * F1 + TB - 1) / TB, TB, 0, stream>>>(out1, b1, N * F1, F1);

    // ============ Layer 2: GATConv(128 -> 1x64) + bias ====================
    gat_wmma_gemm_f32<<<dim3(N / 16, F2 / 64), 128, lds2, stream>>>(out1, W2, h2, F1, F2);
    gat_attn_logits<<<(N + TB - 1) / TB, TB, 0, stream>>>(h2, a2s, a2d,
                                                          al2s, al2d, N, 1, F2);
    gat_edge_max    <<<(Etot + TB - 1) / TB, TB, 0, stream>>>(ei, E, Etot,
                                                              al2s, al2d, mk2, 1);
    gat_edge_denom  <<<(Etot + TB - 1) / TB, TB, 0, stream>>>(ei, E, Etot,
                                                              al2s, al2d, mk2, dn2, 1);
    gat_edge_scatter<<<(Etot + TB - 1) / TB, TB, 0, stream>>>(ei, E, Etot,
                                                              al2s, al2d, mk2, dn2,
                                                              h2, (float*)d_out, 1, F2);
    gat_bias_add<<<((size_t)N * F2 + TB - 1) / TB, TB, 0, stream>>>((float*)d_out, b2,
                                                                    N * F2, F2);
}